// WeightedDTMLayer_76613626626474
// MI455X (gfx1250) — compile-verified
//
#include <hip/hip_runtime.h>
#include <math.h>

// ---------------------------------------------------------------------------
// WeightedDTMLayer for MI455X (gfx1250, wave32).
//   x: (4,3,64,64) f32  ->  out: (4,3,64,64) f32
// Strategy:
//   Kernel 1: build sorted neighbor-offset table via counting sort over
//             key = dx^2+dy^2 (deterministic order: key asc, then dx asc,
//             then +dy before -dy).  Written to d_ws.
//   Kernel 2: per (channel, 256-point tile) block:
//             - TDM tensor_load_to_lds stages the 16KB weight plane to LDS
//             - first 2048 table entries staged to LDS, tail prefetched
//             - channel sum via chained V_WMMA_F32_16X16X4_F32 (B = ones)
//             - per-lane walk over ascending-distance offsets to the
//               bound crossing; dtm = sqrt((S_d + d^2*(b - S_w))/b)
// ---------------------------------------------------------------------------

typedef __attribute__((ext_vector_type(2))) float    v2f;
typedef __attribute__((ext_vector_type(8))) float    v8f;
typedef __attribute__((ext_vector_type(4))) unsigned u32x4;
typedef __attribute__((ext_vector_type(8))) unsigned u32x8;

#define NKEY  7939          // keys 0..7938 (= 2*63^2)
#define NPAIR (127 * 127)   // 16129 offset pairs
#define NTAB  16129
#define TLDS  2048          // table entries staged in LDS

// ===========================================================================
// Kernel 1: deterministic counting sort of neighbor offsets by squared key.
// Entry encoding: bits[6:0]=dx+64, bits[13:7]=dy+64, bits[26:14]=key.
// ===========================================================================
__global__ __launch_bounds__(1024) void dtm_build_table(unsigned* __restrict__ tab) {
  __shared__ int   cnt[8192];   // per-key counts -> exclusive bases
  __shared__ short sq[8192];    // integer-sqrt table: sq[v*v]=v else -1
  __shared__ int   part[1024];  // scan partials

  const int t = threadIdx.x;
  for (int i = t; i < 8192; i += 1024) { cnt[i] = 0; sq[i] = -1; }
  __syncthreads();
  if (t < 90) { int s2 = t * t; if (s2 < 8192) sq[s2] = (short)t; }
  __syncthreads();

  // Histogram of keys (count values are order-independent -> deterministic).
  for (int i = t; i < NPAIR; i += 1024) {
    int dx = i % 127 - 63, dy = i / 127 - 63;
    atomicAdd(&cnt[dx * dx + dy * dy], 1);
  }
  __syncthreads();

  // Exclusive scan over 8192 counts: 8 per thread + Hillis-Steele on partials.
  const int base = t * 8;
  int local[8]; int s = 0;
  for (int j = 0; j < 8; ++j) { local[j] = cnt[base + j]; s += local[j]; }
  part[t] = s;
  __syncthreads();
  for (int off = 1; off < 1024; off <<= 1) {
    int v = (t >= off) ? part[t - off] : 0;
    __syncthreads();
    part[t] += v;
    __syncthreads();
  }
  int run = (t == 0) ? 0 : part[t - 1];
  for (int j = 0; j < 8; ++j) { cnt[base + j] = run; run += local[j]; }
  __syncthreads();

  // Scatter: each thread owns keys t, t+1024, ... ; fixed enumeration order.
  for (int k = t; k < NKEY; k += 1024) {
    int pos = cnt[k];
    for (int dx = -63; dx <= 63; ++dx) {
      int rem = k - dx * dx;
      if (rem < 0) continue;
      int dy = sq[rem];
      if (dy < 0 || dy > 63) continue;
      unsigned hi = (unsigned)(k << 14) | (unsigned)(dx + 64);
      tab[pos++] = hi | (unsigned)((dy + 64) << 7);
      if (dy > 0) tab[pos++] = hi | (unsigned)((-dy + 64) << 7);
    }
  }
}

// ===========================================================================
// Kernel 2: bound via WMMA reduction + DTM walk. grid = 12 channels x 16 tiles.
// ===========================================================================
__global__ __launch_bounds__(256) void dtm_kernel(const float* __restrict__ x,
                                                  const unsigned* __restrict__ tab,
                                                  float* __restrict__ out) {
  __shared__ float    sw[4096];   // 16KB weight plane for this channel
  __shared__ unsigned tb[TLDS];   // 8KB head of sorted offset table
  __shared__ float    wsum[8];

  const int ch   = blockIdx.x >> 4;   // 0..11
  const int tile = blockIdx.x & 15;   // 0..15
  const int t    = threadIdx.x;
  const float* wp = x + ch * 4096;

  // ---- Tensor Data Mover: async DMA of the weight plane into LDS ----------
  {
    unsigned long long ga = (unsigned long long)wp;
    unsigned lds0 = (unsigned)(unsigned long long)(&sw[0]);
    // D# group0: count=1 | lds_addr | global_addr[56:0] | type=2
    u32x4 g0 = { 1u,
                 lds0,
                 (unsigned)ga,
                 ((unsigned)(ga >> 32) & 0x01FFFFFFu) | 0x80000000u };
    // D# group1: data_size=4B; tensor_dim0=4096; tensor_dim1=1;
    //            tile_dim0=4096; tile_dim1=1; dim0_stride=4096; dim1_stride=4096
    u32x8 g1 = { 2u << 16,        // workgroup_mask=0, data_size=2 (4B)
                 4096u << 16,     // tensor_dim0[15:0] @ bit48
                 1u << 16,        // tensor_dim0 hi=0 | tensor_dim1=1
                 4096u << 16,     // tensor_dim1 hi=0 | tile_dim0=4096
                 1u,              // tile_dim1=1, tile_dim2=0
                 4096u,           // tensor_dim0_stride lo
                 4096u << 16,     // dim0_stride hi=0 | tensor_dim1_stride lo16
                 0u };            // tensor_dim1_stride hi
    if (t == 0) {
      asm volatile("tensor_load_to_lds %0, %1" :: "s"(g0), "s"(g1) : "memory");
    }
  }

  // ---- Stage head of offset table; prefetch the (rarely used) global tail --
  {
    const uint4* src = (const uint4*)tab;
    uint4*       dst = (uint4*)tb;
    for (int i = t; i < TLDS / 4; i += 256) dst[i] = src[i];
  }
  __builtin_prefetch(tab + TLDS, 0, 1);   // -> global_prefetch_b8

  __builtin_amdgcn_s_wait_tensorcnt(0);   // TDM completion (per-wave)
  __syncthreads();

  // ---- Channel sum on the matrix core: C += A(16x4 f32) x ones(4x16) ------
  // Each of the 8 waves reduces 512 contiguous weights with 8 chained WMMAs.
  const int wv = t >> 5, ln = t & 31;
  {
    v8f c = {0.f, 0.f, 0.f, 0.f, 0.f, 0.f, 0.f, 0.f};
    const v2f one = {1.0f, 1.0f};
    const int base0 = wv * 512;
    for (int j = 0; j < 8; ++j) {
      int b = base0 + j * 64 + 2 * ln;
      v2f a = { sw[b], sw[b + 1] };
      c = __builtin_amdgcn_wmma_f32_16x16x4_f32(
          /*neg_a=*/false, a, /*neg_b=*/false, one,
          /*c_mod=*/(short)0, c, /*reuse_a=*/false, /*reuse_b=*/false);
    }
    // D[m][n] identical over n; lane L holds rows 0..7 (L<16) / 8..15 (L>=16).
    float s = c[0] + c[1] + c[2] + c[3] + c[4] + c[5] + c[6] + c[7];
    float tw = __shfl(s, 0, 32) + __shfl(s, 16, 32);  // full 512-elem sum
    if (ln == 0) wsum[wv] = tw;
  }
  __syncthreads();
  float total = 0.f;
  for (int i = 0; i < 8; ++i) total += wsum[i];
  const float bound = 0.05f * total;

  // ---- DTM walk: ascending-distance offsets until weight crosses bound ----
  const int p  = tile * 256 + t;        // consecutive lanes -> consecutive px
  const int px = p & 63, py = p >> 6;
  const float hh = (1.0f / 63.0f) * (1.0f / 63.0f);

  float accW = 0.f, accD = 0.f, res = 0.f;
  bool done = false;
  for (int i = 0; i < NTAB && !done; ++i) {
    unsigned e = (i < TLDS) ? tb[i] : tab[i];     // uniform read (broadcast)
    int dx = (int)(e & 127u) - 64;
    int dy = (int)((e >> 7) & 127u) - 64;
    int nx = px + dx, ny = py + dy;
    if ((unsigned)nx < 64u && (unsigned)ny < 64u) {
      float w  = sw[(ny << 6) + nx];              // stride-1 across lanes
      float r2 = (float)(e >> 14) * hh;
      float nw = accW + w;
      if (nw >= bound) {                           // crossing neighbor k
        res = accD + r2 * (bound - accW);
        done = true;
      } else {
        accW = nw;
        accD += r2 * w;
      }
    }
  }
  if (!done) res = accD;                           // unreachable in practice

  out[ch * 4096 + p] = sqrtf(res / bound);
}

// ===========================================================================
extern "C" void kernel_launch(void* const* d_in, const int* in_sizes, int n_in,
                              void* d_out, int out_size, void* d_ws, size_t ws_size,
                              hipStream_t stream) {
  (void)in_sizes; (void)n_in; (void)out_size; (void)ws_size;
  const float* x   = (const float*)d_in[0];
  float*       out = (float*)d_out;
  unsigned*    tab = (unsigned*)d_ws;   // 16129 * 4B sorted offset table

  hipLaunchKernelGGL(dtm_build_table, dim3(1),   dim3(1024), 0, stream, tab);
  hipLaunchKernelGGL(dtm_kernel,      dim3(192), dim3(256),  0, stream, x, tab, out);
}